// INNRotatELinkPredictor_42064909697223
// MI455X (gfx1250) — compile-verified
//
#include <hip/hip_runtime.h>
#include <hip/hip_bf16.h>

#define DIM      128
#define BATCH    32
#define TILE_E   32
#define THREADS  256

#if defined(__has_builtin)
#  if __has_builtin(__builtin_amdgcn_global_load_async_to_lds_b128) && \
      __has_builtin(__builtin_amdgcn_s_wait_asynccnt)
#    define USE_ASYNC_LDS 1
#  endif
#endif
#ifndef USE_ASYNC_LDS
#  define USE_ASYNC_LDS 0
#endif

typedef int v4i __attribute__((ext_vector_type(4)));
#define AS1 __attribute__((address_space(1)))
#define AS3 __attribute__((address_space(3)))

__device__ __forceinline__ float softplus_f(float x) {
  // stable softplus: max(x,0) + log(1 + exp(-|x|))
  return fmaxf(x, 0.0f) + logf(1.0f + expf(-fabsf(x)));
}

// ---------------------------------------------------------------------------
// Kernel 1: per-batch rotated predictions, stored transposed predT[d][b],
// plus pred_rsum[b] = sum_d softplus(ent_rho[h,d]) + softplus(rel_rho[r,d])
// ---------------------------------------------------------------------------
__global__ __launch_bounds__(DIM) void prep_kernel(
    const float* __restrict__ ent_center, const float* __restrict__ ent_rho,
    const float* __restrict__ rel_center, const float* __restrict__ rel_rho,
    const int*   __restrict__ trip,
    float* __restrict__ predT_re, float* __restrict__ predT_im,
    float* __restrict__ pred_rsum)
{
  __shared__ float red[DIM];
  const int b = blockIdx.x;
  const int d = threadIdx.x;
  const size_t h = (size_t)trip[b * 3 + 0];
  const size_t r = (size_t)trip[b * 3 + 1];

  const float hre = ent_center[h * (2 * DIM) + d];
  const float him = ent_center[h * (2 * DIM) + DIM + d];
  const float ph  = rel_center[r * DIM + d];
  const float s = sinf(ph);
  const float c = cosf(ph);

  predT_re[d * BATCH + b] = hre * c - him * s;
  predT_im[d * BATCH + b] = hre * s + him * c;

  red[d] = softplus_f(ent_rho[h * DIM + d]) + softplus_f(rel_rho[r * DIM + d]);
  __syncthreads();
  if (d == 0) {
    float t = 0.0f;
    for (int i = 0; i < DIM; ++i) t += red[i];
    pred_rsum[b] = t;
  }
}

// ---------------------------------------------------------------------------
// Kernel 2: u_rsum[e] = sum_d softplus(ent_rho[e,d]); one wave32 per row
// ---------------------------------------------------------------------------
__global__ __launch_bounds__(256) void ursum_kernel(
    const float* __restrict__ ent_rho, float* __restrict__ u_rsum, int num_ent)
{
  const int row  = blockIdx.x * 8 + (threadIdx.x >> 5);
  const int lane = threadIdx.x & 31;
  if (row >= num_ent) return;
  float s = 0.0f;
#pragma unroll
  for (int j = 0; j < 4; ++j)
    s += softplus_f(ent_rho[(size_t)row * DIM + lane + j * 32]);
#pragma unroll
  for (int off = 16; off > 0; off >>= 1)
    s += __shfl_down(s, off, 32);
  if (lane == 0) u_rsum[row] = s;
}

// ---------------------------------------------------------------------------
// Kernel 3: main distance kernel.
//   block = 256 threads (8 waves), tile = 32 entities.
//   wave w owns entities ebase = 4*w .. 4*w+3 ; lane = batch index b.
//   ent_center tile staged to LDS with async global->LDS copy (CDNA5 path).
// ---------------------------------------------------------------------------
__global__ __launch_bounds__(THREADS) void dist_kernel(
    const float* __restrict__ ent_center,
    const float* __restrict__ predT_re,
    const float* __restrict__ predT_im,
    const float* __restrict__ pred_rsum,
    const float* __restrict__ u_rsum,
    float* __restrict__ out, int num_ent)
{
  __shared__ __align__(16) float u[TILE_E * 2 * DIM];   // 32 KB: [e][re 0..127 | im 0..127]
  __shared__ float otile[BATCH][TILE_E + 1];            // transpose pad: conflict-free

  const int tid = threadIdx.x;
  const int e0  = blockIdx.x * TILE_E;
  const int n_ent_tile = min(TILE_E, num_ent - e0);
  const int nvec = n_ent_tile * (2 * DIM / 4);          // float4 count to stage

  const float* gsrc = ent_center + (size_t)e0 * (2 * DIM);

#if USE_ASYNC_LDS
  for (int i = tid; i < nvec; i += THREADS) {
    __builtin_amdgcn_global_load_async_to_lds_b128(
        (AS1 v4i*)(gsrc + (size_t)i * 4),
        (AS3 v4i*)(u + i * 4),
        0, 0);
  }
  __builtin_amdgcn_s_wait_asynccnt(0);
  __syncthreads();
#else
  {
    const float4* s4 = (const float4*)gsrc;
    float4* d4 = (float4*)u;
    for (int i = tid; i < nvec; i += THREADS) d4[i] = s4[i];
    __syncthreads();
  }
#endif

  const int wid   = tid >> 5;
  const int lane  = tid & 31;      // == batch index b (wave32 == BATCH)
  const int ebase = wid * 4;

  float acc[4] = {0.f, 0.f, 0.f, 0.f};

  for (int c = 0; c < 8; ++c) {    // 8 chunks of 16 dims
    const int d0 = c * 16;
    float pr[16], pi[16];
#pragma unroll
    for (int j = 0; j < 16; ++j) {  // coalesced: lane-consecutive, L2-resident
      pr[j] = predT_re[(d0 + j) * BATCH + lane];
      pi[j] = predT_im[(d0 + j) * BATCH + lane];
    }
#pragma unroll
    for (int k = 0; k < 4; ++k) {
      const float4* ur = (const float4*)(u + (ebase + k) * (2 * DIM) + d0);
      const float4* ui = (const float4*)(u + (ebase + k) * (2 * DIM) + DIM + d0);
      float a = acc[k];
#pragma unroll
      for (int q = 0; q < 4; ++q) {
        const float4 rv = ur[q];   // wave-uniform LDS broadcast (ds_load_b128)
        const float4 iv = ui[q];
        float dr, di;
        dr = pr[q * 4 + 0] - rv.x; di = pi[q * 4 + 0] - iv.x;
        a += __builtin_amdgcn_sqrtf(__builtin_fmaf(dr, dr, di * di));
        dr = pr[q * 4 + 1] - rv.y; di = pi[q * 4 + 1] - iv.y;
        a += __builtin_amdgcn_sqrtf(__builtin_fmaf(dr, dr, di * di));
        dr = pr[q * 4 + 2] - rv.z; di = pi[q * 4 + 2] - iv.z;
        a += __builtin_amdgcn_sqrtf(__builtin_fmaf(dr, dr, di * di));
        dr = pr[q * 4 + 3] - rv.w; di = pi[q * 4 + 3] - iv.w;
        a += __builtin_amdgcn_sqrtf(__builtin_fmaf(dr, dr, di * di));
      }
      acc[k] = a;
    }
  }

  // transpose through LDS so global stores are coalesced along entities
#pragma unroll
  for (int k = 0; k < 4; ++k) otile[lane][ebase + k] = acc[k];
  __syncthreads();

  const int b  = tid >> 3;          // 0..31
  const int ec = (tid & 7) * 4;     // 0..28
  const float ps = pred_rsum[b];
#pragma unroll
  for (int k = 0; k < 4; ++k) {
    const int e = e0 + ec + k;
    if (e < num_ent)
      out[(size_t)b * num_ent + e] = ps + u_rsum[e] - otile[b][ec + k];
  }
}

// ---------------------------------------------------------------------------
extern "C" void kernel_launch(void* const* d_in, const int* in_sizes, int n_in,
                              void* d_out, int out_size, void* d_ws, size_t ws_size,
                              hipStream_t stream) {
  const float* ent_center = (const float*)d_in[0];
  const float* ent_rho    = (const float*)d_in[1];
  const float* rel_center = (const float*)d_in[2];
  const float* rel_rho    = (const float*)d_in[3];
  const int*   trip       = (const int*)d_in[4];
  float* out = (float*)d_out;

  const int num_ent = in_sizes[1] / DIM;   // 50000
  const int batch   = in_sizes[4] / 3;     // 32

  float* ws        = (float*)d_ws;
  float* predT_re  = ws;                                   // DIM*BATCH
  float* predT_im  = ws + DIM * BATCH;                     // DIM*BATCH
  float* pred_rsum = ws + 2 * DIM * BATCH;                 // BATCH
  float* u_rsum    = ws + 2 * DIM * BATCH + BATCH;         // num_ent

  prep_kernel<<<batch, DIM, 0, stream>>>(ent_center, ent_rho, rel_center,
                                         rel_rho, trip, predT_re, predT_im,
                                         pred_rsum);
  ursum_kernel<<<(num_ent + 7) / 8, 256, 0, stream>>>(ent_rho, u_rsum, num_ent);
  dist_kernel<<<(num_ent + TILE_E - 1) / TILE_E, THREADS, 0, stream>>>(
      ent_center, predT_re, predT_im, pred_rsum, u_rsum, out, num_ent);
}